// InferenceTransform_66202625900988
// MI455X (gfx1250) — compile-verified
//
#include <hip/hip_runtime.h>
#include <math.h>

// ---------------------------------------------------------------------------
// RetinaNet post-process for MI455X (gfx1250, wave32).
//
// Roofline: dominated by streaming 800 MB of classification scores once
// (4 x 100000 x 500 fp32).  23.3 TB/s HBM => ~34 us floor.  Pure bandwidth
// problem; no GEMM structure, so no WMMA.  We use the CDNA5 async-copy path
// (global_load_async_to_lds_b128 + s_wait_asynccnt) to double-buffer row
// tiles into LDS, and nontemporal loads in the fallback so the once-touched
// stream does not thrash the 192 MB L2.
// ---------------------------------------------------------------------------

#define BB 4
#define NN 100000
#define CC 500
#define ROWS (BB * NN)
#define CHUNKS 125              // float4 chunks per row (500 floats, 16B aligned rows)
#define WAVES_PER_BLOCK 8
#define THREADS (WAVES_PER_BLOCK * 32)
#define NBLOCKS 4096            // 32768 waves -> ~12 rows/wave (pipelines the async copies)

typedef float v4f __attribute__((ext_vector_type(4)));
typedef int   v4i __attribute__((ext_vector_type(4)));

#if defined(__AMDGCN__) && __has_builtin(__builtin_amdgcn_global_load_async_to_lds_b128)
#define USE_ASYNC 1
// Builtin expects AS(1) int4* source (per hipcc diagnostic) and an LDS-side
// pointer; give it address-space-qualified int4 pointers for both.
typedef __attribute__((address_space(1))) v4i v4i_g;
typedef __attribute__((address_space(3))) v4i v4i_l;
#else
#define USE_ASYNC 0
#endif

__global__ __launch_bounds__(THREADS) void retina_post_kernel(
    const float* __restrict__ cls,   // [B,N,C]
    const float* __restrict__ reg,   // [B,N,4]
    const float* __restrict__ anc,   // [1,N,4]
    const float* __restrict__ thr,   // [1]
    const float* __restrict__ rf,    // [4]
    float* __restrict__ out)         // boxes | cls_idx | scores | mask (flat)
{
#if USE_ASYNC
    __shared__ float lds[WAVES_PER_BLOCK][2][512];   // 2 x 2KB slots per wave = 32KB/block
#endif
    const int lane = threadIdx.x & 31;
    const int wid  = threadIdx.x >> 5;
    const int gw   = blockIdx.x * WAVES_PER_BLOCK + wid;   // global wave id
    const int GW   = gridDim.x * WAVES_PER_BLOCK;

    float* out_boxes = out;
    float* out_cls   = out + (size_t)ROWS * 4;
    float* out_sc    = out + (size_t)ROWS * 5;
    float* out_mask  = out + (size_t)ROWS * 6;

    const float t   = thr[0];
    const float rf0 = rf[0], rf1 = rf[1], rf2 = rf[2], rf3 = rf[3];

#if USE_ASYNC
    // Stage one 2000B row into an LDS slot via the CDNA5 async copy engine.
    auto stage = [&](int row, int slot) {
        const float* src = cls + (size_t)row * CC;
#pragma unroll
        for (int k = 0; k < 4; ++k) {
            int idx = k * 32 + lane;
            if (idx < CHUNKS) {
                __builtin_amdgcn_global_load_async_to_lds_b128(
                    (v4i_g*)(src + idx * 4),
                    (v4i_l*)&lds[wid][slot][idx * 4],
                    /*offset=*/0, /*cpol=*/0);
            }
        }
    };
#endif

    int row = gw;
#if USE_ASYNC
    int slot = 0;
    if (row < ROWS) stage(row, 0);
#endif

    for (; row < ROWS; row += GW) {
#if USE_ASYNC
        const int nxt = row + GW;
        if (nxt < ROWS) {
            stage(nxt, slot ^ 1);                       // prefetch next row
            asm volatile("s_wait_asynccnt 4" ::: "memory");  // current slot's 4 copies done
        } else {
            asm volatile("s_wait_asynccnt 0" ::: "memory");
        }
        const v4f* p = (const v4f*)&lds[wid][slot][0];
#else
        const v4f* p = (const v4f*)(cls + (size_t)row * CC);
#endif
        // ---- per-lane max/argmax over this lane's float4 chunks ----
        float m = -3.402823466e38f;
        int   mi = 0;
#pragma unroll
        for (int k = 0; k < 4; ++k) {
            int j = k * 32 + lane;
            if (j < CHUNKS) {
#if USE_ASYNC
                v4f v = p[j];                                   // ds_load_b128
#else
                v4f v = __builtin_nontemporal_load(&p[j]);      // global_load_b128 th:NT
#endif
                int base = j * 4;
                if (v.x > m) { m = v.x; mi = base;     }
                if (v.y > m) { m = v.y; mi = base + 1; }
                if (v.z > m) { m = v.z; mi = base + 2; }
                if (v.w > m) { m = v.w; mi = base + 3; }
            }
        }

        // ---- wave32 reduction (max, first-occurrence index) ----
#pragma unroll
        for (int off = 16; off > 0; off >>= 1) {
            float om  = __shfl_xor(m, off, 32);
            int   omi = __shfl_xor(mi, off, 32);
            if (om > m || (om == m && omi < mi)) { m = om; mi = omi; }
        }

        // ---- box decode + masked writeback (lane 0) ----
        if (lane == 0) {
            const int n = row % NN;
            v4f a = *(const v4f*)(anc + (size_t)n * 4);
            v4f r = *(const v4f*)(reg + (size_t)row * 4);

            float fw = a.z - a.x;
            float fh = a.w - a.y;
            float cx = a.x + 0.5f * fw;
            float cy = a.y + 0.5f * fh;

            float pcx = cx + (r.x * rf0) * fw;
            float pcy = cy + (r.y * rf1) * fh;
            float pw  = __expf(r.z * rf2) * fw;
            float ph  = __expf(r.w * rf3) * fh;

            float x1 = pcx - 0.5f * pw, y1 = pcy - 0.5f * ph;
            float x2 = pcx + 0.5f * pw, y2 = pcy + 0.5f * ph;
            x1 = fminf(fmaxf(x1, 0.f), 512.f);
            x2 = fminf(fmaxf(x2, 0.f), 512.f);
            y1 = fminf(fmaxf(y1, 0.f), 512.f);
            y2 = fminf(fmaxf(y2, 0.f), 512.f);

            const float fm = (m > t) ? 1.f : 0.f;
            v4f box = { x1 * fm, y1 * fm, x2 * fm, y2 * fm };
            *(v4f*)(out_boxes + (size_t)row * 4) = box;
            out_cls[row]  = fm * (float)mi;
            out_sc[row]   = fm * m;
            out_mask[row] = fm;
        }
#if USE_ASYNC
        slot ^= 1;
#endif
    }
}

extern "C" void kernel_launch(void* const* d_in, const int* in_sizes, int n_in,
                              void* d_out, int out_size, void* d_ws, size_t ws_size,
                              hipStream_t stream) {
    // setup_inputs order: imgs(0, unused), classifications(1), regressions(2),
    //                     anchors(3), cls_thresh(4), regress_factor(5)
    const float* cls = (const float*)d_in[1];
    const float* reg = (const float*)d_in[2];
    const float* anc = (const float*)d_in[3];
    const float* thr = (const float*)d_in[4];
    const float* rf  = (const float*)d_in[5];
    float* out = (float*)d_out;

    hipLaunchKernelGGL(retina_post_kernel, dim3(NBLOCKS), dim3(THREADS), 0, stream,
                       cls, reg, anc, thr, rf, out);
}